// StabilizedButterflyLayer_17111149707329
// MI455X (gfx1250) — compile-verified
//
#include <hip/hip_runtime.h>
#include <math.h>

// ---------------------------------------------------------------------------
// StabilizedButterflyLayer fused kernel for MI455X (gfx1250, wave32)
//
// B=2048 rows, n=4096 features, 12 stages. Fully fused: row data lives in
// registers across all 12 stages (rotation -> LayerNorm -> block scale).
// HBM traffic = read x + write out = 64MB -> ~2.7us roofline @ 23.3 TB/s.
//
// CDNA5-specific paths used:
//   * tensor_load_to_lds (TDM) + s_wait_tensorcnt for the input tile
//   * v_wmma_f32_16x16x4_f32 for the 32-lane wave reductions (matrix pipe
//     offload; B = all-ones makes the result independent of B's VGPR layout)
//   * global_prefetch for next-stage parameters
// ---------------------------------------------------------------------------

typedef float v2f  __attribute__((ext_vector_type(2)));
typedef float v8f  __attribute__((ext_vector_type(8)));
typedef unsigned int u32x4 __attribute__((ext_vector_type(4)));
typedef int  i32x4 __attribute__((ext_vector_type(4)));
typedef int  i32x8 __attribute__((ext_vector_type(8)));

#define N_FEAT   4096
#define NPAIRS   2048
#define NBLOCKS  256          // 4096 / 16
#define STAGES   12
#define RPB      4            // rows per block
#define TPB      256          // threads per block (8 wave32s)

// ---------------------------------------------------------------------------
// Prep: cos/sin tables, and gamma/beta with block_scales folded in.
//   cs[st][p][0..1] = {cos, sin}      (12 x 2048 x 2 floats)
//   gs[st][i] = gamma[st][i] * bscale[st][i/16]
//   bs[st][i] = beta [st][i] * bscale[st][i/16]
// ---------------------------------------------------------------------------
__global__ __launch_bounds__(TPB) void bf_prep(
    const float* __restrict__ ang, const float* __restrict__ gam,
    const float* __restrict__ bet, const float* __restrict__ bsc,
    float* __restrict__ cs, float* __restrict__ gs, float* __restrict__ bs) {
  int idx = blockIdx.x * TPB + threadIdx.x;
  if (idx >= STAGES * N_FEAT) return;
  int st = idx >> 12;             // / 4096
  int i  = idx & (N_FEAT - 1);
  float sc = bsc[st * NBLOCKS + (i >> 4)];
  gs[idx] = gam[idx] * sc;
  bs[idx] = bet[idx] * sc;
  if (i < NPAIRS) {
    float th = ang[st * NPAIRS + i];
    cs[(st * NPAIRS + i) * 2 + 0] = cosf(th);
    cs[(st * NPAIRS + i) * 2 + 1] = sinf(th);
  }
}

// ---------------------------------------------------------------------------
// 32-lane sum via V_WMMA_F32_16X16X4_F32.
// A VGPR0 = x (per-lane value), A VGPR1 = 0  => A[m][0]=x_m, A[m][2]=x_{m+16}.
// B = all ones (both VGPRs = 1.0 in every lane -> layout independent).
// D[m][n] = x_m + x_{m+16}.  Sum of the 8 D VGPRs gives the low-half total in
// lanes 0-15 and the high-half total in lanes 16-31; one xor-16 shuffle adds
// the halves so every lane holds the full 32-lane sum.
// ---------------------------------------------------------------------------
__device__ __forceinline__ float wave_sum32(float x) {
  v2f a; a.x = x;    a.y = 0.0f;
  v2f b; b.x = 1.0f; b.y = 1.0f;
  v8f c = {};
  v8f d = __builtin_amdgcn_wmma_f32_16x16x4_f32(
      false, a, false, b, (short)0, c, false, false);
  float p = ((d[0] + d[1]) + (d[2] + d[3])) + ((d[4] + d[5]) + (d[6] + d[7]));
  p += __shfl_xor(p, 16, 32);
  return p;
}

// ---------------------------------------------------------------------------
// Main fused kernel: one block = 4 rows, 12 stages in registers.
// Thread t owns columns [16t, 16t+16) of each of its 4 rows.
// ---------------------------------------------------------------------------
__global__ __launch_bounds__(TPB) void bf_main(
    const float* __restrict__ x,  const float* __restrict__ cs,
    const float* __restrict__ gs, const float* __restrict__ bs,
    float* __restrict__ out) {
  __shared__ __align__(16) float xs[RPB * N_FEAT];       // 64 KB input tile
  __shared__ __align__(16) float red[2][RPB][16];        // S[0..7] | Q[0..7]

  const int tid  = threadIdx.x;
  const int wave = tid >> 5;
  const int lane = tid & 31;
  const size_t row0 = (size_t)blockIdx.x * RPB;

  // ---- TDM: async 2D tile load (4 rows x 4096 f32) into LDS, wave 0 only --
  if (wave == 0) {
    unsigned long long ga = (unsigned long long)(uintptr_t)(x + row0 * N_FEAT);
    unsigned int la = (unsigned int)(uintptr_t)(&xs[0]);   // LDS byte offset

    u32x4 g0;
    g0.x = 1u;                                             // count = 1 (valid)
    g0.y = la;                                             // lds_addr
    g0.z = (unsigned int)(ga & 0xffffffffull);             // global_addr[31:0]
    g0.w = (unsigned int)((ga >> 32) & 0x01ffffffull)      // global_addr[56:32]
         | (2u << 30);                                     // type = 2 (image)

    i32x8 g1;
    g1[0] = 0x00020000;                    // data_size = 2 (4 bytes)
    g1[1] = (int)(4096u << 16);            // tensor_dim0[15:0] = 4096
    g1[2] = (int)((unsigned)RPB << 16);    // tensor_dim1[15:0] = RPB
    g1[3] = (int)(4096u << 16);            // tile_dim0 = 4096
    g1[4] = RPB;                           // tile_dim1 = RPB
    g1[5] = 4096;                          // tensor_dim0_stride[31:0]
    g1[6] = 0;
    g1[7] = 0;
    i32x4 z4 = {0, 0, 0, 0};
#if defined(__clang_major__) && (__clang_major__ >= 23)
    i32x8 z8 = {0, 0, 0, 0, 0, 0, 0, 0};
    __builtin_amdgcn_tensor_load_to_lds(g0, g1, z4, z4, z8, 0);
#else
    __builtin_amdgcn_tensor_load_to_lds(g0, g1, z4, z4, 0);
#endif
    __builtin_amdgcn_s_wait_tensorcnt(0);
  }
  __syncthreads();

  // ---- pull this thread's 4x16 slice into registers ----
  float v[RPB][16];
#pragma unroll
  for (int r = 0; r < RPB; ++r) {
    const float4* src = (const float4*)&xs[r * N_FEAT + tid * 16];
#pragma unroll
    for (int k = 0; k < 4; ++k) {
      float4 t = src[k];
      v[r][4*k+0] = t.x; v[r][4*k+1] = t.y;
      v[r][4*k+2] = t.z; v[r][4*k+3] = t.w;
    }
  }

  const float inv_n = 1.0f / (float)N_FEAT;

  for (int st = 0; st < STAGES; ++st) {
    const int p = st & 1;
    const float4* csv = (const float4*)(cs + (size_t)st * (2 * NPAIRS) + tid * 16);
    const float4* gv  = (const float4*)(gs + (size_t)st * N_FEAT + tid * 16);
    const float4* bv  = (const float4*)(bs + (size_t)st * N_FEAT + tid * 16);

    float cc[8], ss[8], g[16], b[16];
#pragma unroll
    for (int k = 0; k < 4; ++k) {
      float4 t = csv[k];
      cc[2*k] = t.x; ss[2*k] = t.y; cc[2*k+1] = t.z; ss[2*k+1] = t.w;
      float4 tg = gv[k];
      g[4*k] = tg.x; g[4*k+1] = tg.y; g[4*k+2] = tg.z; g[4*k+3] = tg.w;
      float4 tb = bv[k];
      b[4*k] = tb.x; b[4*k+1] = tb.y; b[4*k+2] = tb.z; b[4*k+3] = tb.w;
    }

    if (st + 1 < STAGES) {  // warm L0 for next stage's (L2-resident) params
      __builtin_prefetch(gs + (size_t)(st + 1) * N_FEAT + tid * 16, 0, 1);
      __builtin_prefetch(bs + (size_t)(st + 1) * N_FEAT + tid * 16, 0, 1);
      __builtin_prefetch(cs + (size_t)(st + 1) * (2 * NPAIRS) + tid * 16, 0, 1);
    }

    // ---- rotate pairs + accumulate sum / sumsq, wave-reduce via WMMA ----
#pragma unroll
    for (int r = 0; r < RPB; ++r) {
      float sA = 0.0f, qA = 0.0f;
#pragma unroll
      for (int j = 0; j < 8; ++j) {
        float x0 = v[r][2*j], x1 = v[r][2*j+1];
        float y0 = fmaf(-ss[j], x1, cc[j] * x0);
        float y1 = fmaf( ss[j], x0, cc[j] * x1);
        v[r][2*j] = y0; v[r][2*j+1] = y1;
        sA += y0 + y1;
        qA = fmaf(y0, y0, qA);
        qA = fmaf(y1, y1, qA);
      }
      float S = wave_sum32(sA);
      float Q = wave_sum32(qA);
      if (lane == 0) {
        red[p][r][wave]     = S;
        red[p][r][8 + wave] = Q;
      }
    }
    __syncthreads();   // single barrier per stage (red is double-buffered)

    // ---- finish cross-wave reduction, normalize + (folded) block scale ----
#pragma unroll
    for (int r = 0; r < RPB; ++r) {
      float4 s0 = *(const float4*)&red[p][r][0];
      float4 s1 = *(const float4*)&red[p][r][4];
      float4 q0 = *(const float4*)&red[p][r][8];
      float4 q1 = *(const float4*)&red[p][r][12];
      float S = ((s0.x + s0.y) + (s0.z + s0.w)) + ((s1.x + s1.y) + (s1.z + s1.w));
      float Q = ((q0.x + q0.y) + (q0.z + q0.w)) + ((q1.x + q1.y) + (q1.z + q1.w));
      float mu   = S * inv_n;
      float var  = fmaf(-mu, mu, Q * inv_n);      // biased variance
      float rstd = rsqrtf(var + 1e-5f);
#pragma unroll
      for (int i = 0; i < 16; ++i) {
        float t = rstd * g[i];                    // g,b already carry bscale
        v[r][i] = fmaf(v[r][i], t, fmaf(-mu, t, b[i]));
      }
    }
  }

  // ---- coalesced b128 stores straight from registers ----
#pragma unroll
  for (int r = 0; r < RPB; ++r) {
    float4* dst = (float4*)(out + (row0 + r) * N_FEAT + tid * 16);
#pragma unroll
    for (int k = 0; k < 4; ++k) {
      dst[k] = make_float4(v[r][4*k], v[r][4*k+1], v[r][4*k+2], v[r][4*k+3]);
    }
  }
}

// ---------------------------------------------------------------------------
extern "C" void kernel_launch(void* const* d_in, const int* in_sizes, int n_in,
                              void* d_out, int out_size, void* d_ws, size_t ws_size,
                              hipStream_t stream) {
  (void)n_in; (void)out_size; (void)ws_size;
  const float* x   = (const float*)d_in[0];   // (2048, 4096)
  const float* ang = (const float*)d_in[1];   // (12, 2048)
  const float* gam = (const float*)d_in[2];   // (12, 4096)
  const float* bet = (const float*)d_in[3];   // (12, 4096)
  const float* bsc = (const float*)d_in[4];   // (12, 256)
  float* out = (float*)d_out;

  float* ws = (float*)d_ws;                   // ~576 KB of scratch used
  float* cs = ws;                             // 12*2048*2 floats
  float* gs = cs + STAGES * 2 * NPAIRS;       // 12*4096 floats
  float* bs = gs + STAGES * N_FEAT;           // 12*4096 floats

  int total = STAGES * N_FEAT;
  bf_prep<<<(total + TPB - 1) / TPB, TPB, 0, stream>>>(ang, gam, bet, bsc,
                                                       cs, gs, bs);

  int rows = in_sizes[0] / N_FEAT;            // 2048
  bf_main<<<rows / RPB, TPB, 0, stream>>>(x, cs, gs, bs, out);
}